// SchNetClassifier_88880053223891
// MI455X (gfx1250) — compile-verified
//
#include <hip/hip_runtime.h>
#include <math.h>

typedef __attribute__((ext_vector_type(2))) float v2f;
typedef __attribute__((ext_vector_type(8))) float v8f;

#define GAMMA_F   10.0f
#define NATOMS    512
#define NBATCH    8
#define NF        16
#define ND        20
#define RSTRIDE   32   // padded row stride for R in workspace

// ---------------------------------------------------------------------------
// Kernel A: per batch b, R[b][k] = sum_{m=1..511} exp(-gamma*(||c_b0-c_bm||-mu_k)^2)
// Also zero-fills pad rows 8..15 so kernel B can load A-fragments
// unconditionally (workspace is poisoned, never re-poisoned -> write every call).
// ---------------------------------------------------------------------------
__global__ void schnet_rbf_rowsum(const float* __restrict__ coords,
                                  float* __restrict__ R) {
  const int b   = blockIdx.x;
  const int tid = threadIdx.x;

  __shared__ float sR[ND];
  if (tid < ND) sR[tid] = 0.0f;
  __syncthreads();

  const float cx = coords[(b * NATOMS + 0) * 3 + 0];
  const float cy = coords[(b * NATOMS + 0) * 3 + 1];
  const float cz = coords[(b * NATOMS + 0) * 3 + 2];

  float acc[ND];
#pragma unroll
  for (int k = 0; k < ND; ++k) acc[k] = 0.0f;

  for (int m = tid; m < NATOMS; m += blockDim.x) {
    if (m == 0) continue;  // diagonal mask for row n==0
    const float dx = cx - coords[(b * NATOMS + m) * 3 + 0];
    const float dy = cy - coords[(b * NATOMS + m) * 3 + 1];
    const float dz = cz - coords[(b * NATOMS + m) * 3 + 2];
    const float d2 = dx * dx + dy * dy + dz * dz;
    const float adj = (d2 > 0.0f) ? sqrtf(d2) : 0.0f;
#pragma unroll
    for (int k = 0; k < ND; ++k) {
      const float t = adj - 0.1f * (float)k;
      acc[k] += expf(-GAMMA_F * t * t);
    }
  }

  // wave32 butterfly reduce, then one LDS atomic per wave per bin
#pragma unroll
  for (int k = 0; k < ND; ++k) {
    float v = acc[k];
    for (int off = 16; off > 0; off >>= 1) v += __shfl_xor(v, off, 32);
    if ((tid & 31) == 0) atomicAdd(&sR[k], v);
  }
  __syncthreads();

  if (tid < ND) {
    R[b * RSTRIDE + tid]            = sR[tid];   // real batch rows 0..7
    R[(b + NBATCH) * RSTRIDE + tid] = 0.0f;      // zero pad rows 8..15
  }
}

// ---------------------------------------------------------------------------
// Kernel B: single wave32.  D = A(16x20: batches 0..7 + zero rows) x
// B(20x16 = Wc_w^T) via 5 chained V_WMMA_F32_16X16X4_F32, then epilogue:
//   S[b,f]  = D[b,f] + 511*Wc_b[f]
//   x[f]    = Wi_b[f] + sum_g atom_emb[g]*Wi_w[f,g]
//   v0[b,f] = atom_emb[f] + x[f]*S[b,f]
//   out[b]  = sum_f v0[b,f]*Q_w[f] + Q_b
// ---------------------------------------------------------------------------
__global__ void schnet_wmma_head(const float* __restrict__ R,
                                 const float* __restrict__ atom_emb,
                                 const float* __restrict__ Wc_w,   // (F=16, D=20) row-major
                                 const float* __restrict__ Wc_b,   // (16)
                                 const float* __restrict__ Wi_w,   // (16,16) row-major
                                 const float* __restrict__ Wi_b,   // (16)
                                 const float* __restrict__ Q_w,    // (1,16)
                                 const float* __restrict__ Q_b,    // (1)
                                 float* __restrict__ out) {        // (8)
  const int lane = threadIdx.x;   // 0..31
  const int half = lane >> 4;     // K-half: lanes 0-15 -> K pair {0,1}, 16-31 -> {2,3}
  const int col  = lane & 15;     // A: row M (batch); B: col N (feature)

  v8f acc = {};
#pragma unroll
  for (int c = 0; c < 5; ++c) {   // K = 5 * 4 = 20, exact fit
    const int k0 = 4 * c + 2 * half;
    // A[M=col][K=k0,k0+1]: rows 8..15 were zero-filled in kernel A ->
    // unconditional 8-byte vector load, no EXEC masking needed.
    const v2f a  = *(const v2f*)(R    + col * RSTRIDE + k0);
    // B[K=k0,k0+1][N=col] = Wc_w[col][k0..k0+1]  (B = Wc_w^T), 8B aligned.
    const v2f bb = *(const v2f*)(Wc_w + col * ND      + k0);
    acc = __builtin_amdgcn_wmma_f32_16x16x4_f32(
        /*neg_a=*/false, a, /*neg_b=*/false, bb,
        /*c_mod=*/(short)0, acc, /*reuse_a=*/false, /*reuse_b=*/false);
  }
  // C/D layout: acc[r] on lane L = D[M = r + 8*(L>>4)][N = L&15]
  // -> lanes 0..15 hold batches 0..7 in acc[0..7]; lanes 16..31 hold zero rows.

  const int f     = col;
  const float emb = atom_emb[f];
  const float qw  = Q_w[f];
  const float wcb = Wc_b[f];
  float xf = Wi_b[f];
#pragma unroll
  for (int g = 0; g < NF; ++g) xf += atom_emb[g] * Wi_w[f * NF + g];

  const float qb = Q_b[0];
#pragma unroll
  for (int b = 0; b < NBATCH; ++b) {
    const float S = acc[b] + 511.0f * wcb;   // bias added for all 511 unmasked m
    const float v = emb + xf * S;
    float contrib = (lane < 16) ? v * qw : 0.0f;
    for (int off = 16; off > 0; off >>= 1) contrib += __shfl_xor(contrib, off, 32);
    if (lane == 0) out[b] = contrib + qb;
  }
}

// ---------------------------------------------------------------------------
extern "C" void kernel_launch(void* const* d_in, const int* in_sizes, int n_in,
                              void* d_out, int out_size, void* d_ws, size_t ws_size,
                              hipStream_t stream) {
  const float* coords   = (const float*)d_in[0];  // (8,512,3)
  const float* atom_emb = (const float*)d_in[1];  // (1,1,16)
  const float* Wc_w     = (const float*)d_in[2];  // (16,20)
  const float* Wc_b     = (const float*)d_in[3];  // (16)
  const float* Wi_w     = (const float*)d_in[4];  // (16,16)
  const float* Wi_b     = (const float*)d_in[5];  // (16)
  const float* Q_w      = (const float*)d_in[6];  // (1,16)
  const float* Q_b      = (const float*)d_in[7];  // (1)
  float* out = (float*)d_out;                     // (8,1) fp32
  float* R   = (float*)d_ws;                      // 16*32 floats scratch

  schnet_rbf_rowsum<<<NBATCH, 256, 0, stream>>>(coords, R);
  schnet_wmma_head<<<1, 32, 0, stream>>>(R, atom_emb, Wc_w, Wc_b,
                                         Wi_w, Wi_b, Q_w, Q_b, out);
}